// FocalLoss_46815143526561
// MI455X (gfx1250) — compile-verified
//
#include <hip/hip_runtime.h>
#include <hip/hip_bf16.h>
#include <stdint.h>

#define ALPHA_F 0.25f

// ---------------------------------------------------------------------------
// CDNA5 async global->LDS staging (gfx1250). Guarded: fall back to plain LDS
// fill if the toolchain doesn't expose the builtins.
// Signature (from hipcc diagnostic): (AS1 int*, AS3 int*, imm offset, imm cpol)
// ---------------------------------------------------------------------------
#if defined(__AMDGCN__) && __has_builtin(__builtin_amdgcn_global_load_async_to_lds_b32) && \
    __has_builtin(__builtin_amdgcn_s_wait_asynccnt)
#define HAVE_ASYNC_LDS 1
typedef __attribute__((address_space(1))) int as1_int;
typedef __attribute__((address_space(3))) int as3_int;
#else
#define HAVE_ASYNC_LDS 0
#endif

__device__ __forceinline__ float block_reduce_sum(float v, float* sbuf) {
    // wave32 reduction then cross-wave via LDS; result valid in thread 0.
    #pragma unroll
    for (int off = 16; off > 0; off >>= 1) v += __shfl_down(v, off, 32);
    const int lane = threadIdx.x & 31;
    const int wid  = threadIdx.x >> 5;
    if (lane == 0) sbuf[wid] = v;
    __syncthreads();
    float r = 0.f;
    if (threadIdx.x == 0) {
        const int nw = (blockDim.x + 31) >> 5;
        for (int w = 0; w < nw; ++w) r += sbuf[w];
    }
    __syncthreads();   // allow sbuf reuse by caller
    return r;
}

// focal term for one element; exactly ONE logf per element.
__device__ __forceinline__ float focal_term(float x, bool is_one) {
    float p  = fminf(fmaxf(x, 1e-4f), 1.0f - 1e-4f);
    float w  = is_one ? (1.0f - p) : p;          // focal weight base
    float af = is_one ? ALPHA_F : (1.0f - ALPHA_F);
    float l  = is_one ? p : (1.0f - p);          // log argument
    return af * w * w * (-logf(l));
}

// ---------------------------------------------------------------------------
// init: zero accumulators, count valid annotations per image
// accums layout: [0,B) cls_sum | [B,2B) reg_sum | [2B,3B) num_pos (as float)
// ---------------------------------------------------------------------------
__global__ void init_kernel(const float* __restrict__ anns,
                            float* __restrict__ accums,
                            unsigned* __restrict__ nvalid,
                            int B, int M) {
    __shared__ unsigned cnt;
    const int img = blockIdx.x;
    if (threadIdx.x == 0) cnt = 0;
    __syncthreads();
    unsigned local = 0;
    for (int j = threadIdx.x; j < M; j += blockDim.x)
        if (anns[(size_t)img * M * 5 + (size_t)j * 5 + 4] != -1.0f) local++;
    if (local) atomicAdd(&cnt, local);
    __syncthreads();
    if (threadIdx.x == 0) {
        nvalid[img] = cnt;
        accums[img]         = 0.0f;
        accums[B + img]     = 0.0f;
        accums[2 * B + img] = 0.0f;
    }
}

// ---------------------------------------------------------------------------
// assignment + regression loss: one thread per anchor
// code: >=0 => positive with that label; -1 => negative; -2 => ignore
// ---------------------------------------------------------------------------
__global__ void assign_kernel(const float* __restrict__ anchors,   // [A,4]
                              const float* __restrict__ regs,      // [B,A,4]
                              const float* __restrict__ anns,      // [B,M,5]
                              int* __restrict__ codes,             // [B,A]
                              float* __restrict__ accums,
                              int A, int M, int B) {
    __shared__ float sAnn[5 * 128];      // supports M <= 128
    __shared__ float sred[8];
    const int img = blockIdx.y;
    const int a   = blockIdx.x * blockDim.x + threadIdx.x;
    const float* ann = anns + (size_t)img * M * 5;
    const bool use_lds = (M <= 128);
    if (use_lds) {
        for (int i = threadIdx.x; i < M * 5; i += blockDim.x) sAnn[i] = ann[i];
    }
    __syncthreads();
    const float* annp = use_lds ? sAnn : ann;

    float regsq = 0.0f, poscnt = 0.0f;
    if (a < A) {
        const float ax1 = anchors[a * 4 + 0];
        const float ay1 = anchors[a * 4 + 1];
        const float ax2 = anchors[a * 4 + 2];
        const float ay2 = anchors[a * 4 + 3];
        const float aw  = ax2 - ax1, ah = ay2 - ay1;
        const float areaA = aw * ah;

        float best = -3.0e38f;
        int   arg  = 0;
        for (int m = 0; m < M; ++m) {
            const float bx1 = annp[m * 5 + 0], by1 = annp[m * 5 + 1];
            const float bx2 = annp[m * 5 + 2], by2 = annp[m * 5 + 3];
            const float lab = annp[m * 5 + 4];
            float iw = fminf(ax2, bx2) - fmaxf(ax1, bx1); iw = fmaxf(iw, 0.0f);
            float ih = fminf(ay2, by2) - fmaxf(ay1, by1); ih = fmaxf(ih, 0.0f);
            const float inter = iw * ih;
            const float ua = fmaxf(areaA + (bx2 - bx1) * (by2 - by1) - inter, 1e-8f);
            float iou = inter / ua;
            if (lab == -1.0f) iou = -1.0f;        // masked (invalid annotation)
            if (iou > best) { best = iou; arg = m; }   // first-max wins ties
        }
        const bool pos = (best >= 0.5f);
        const bool neg = (best < 0.4f);
        const int  lab = (int)annp[arg * 5 + 4];
        codes[(size_t)img * A + a] = pos ? lab : (neg ? -1 : -2);

        if (pos) {
            const float gx1 = annp[arg * 5 + 0], gy1 = annp[arg * 5 + 1];
            const float gx2 = annp[arg * 5 + 2], gy2 = annp[arg * 5 + 3];
            const float acx = ax1 + 0.5f * aw, acy = ay1 + 0.5f * ah;
            const float gwr = gx2 - gx1, ghr = gy2 - gy1;
            const float gcx = gx1 + 0.5f * gwr, gcy = gy1 + 0.5f * ghr;
            const float gw = fmaxf(gwr, 1.0f), gh = fmaxf(ghr, 1.0f);
            const float t0 = (gcx - acx) / aw * 10.0f;   // /0.1
            const float t1 = (gcy - acy) / ah * 10.0f;
            const float t2 = logf(gw / aw) * 5.0f;       // /0.2
            const float t3 = logf(gh / ah) * 5.0f;
            const float* r = regs + ((size_t)img * A + a) * 4;
            const float d0 = r[0] - t0, d1 = r[1] - t1, d2 = r[2] - t2, d3 = r[3] - t3;
            regsq  = d0 * d0 + d1 * d1 + d2 * d2 + d3 * d3;
            poscnt = 1.0f;
        }
    }
    float rs = block_reduce_sum(regsq,  sred);
    float ps = block_reduce_sum(poscnt, sred);
    if (threadIdx.x == 0) {
        if (rs != 0.0f) atomicAdd(&accums[B + img], rs);
        if (ps != 0.0f) atomicAdd(&accums[2 * B + img], ps);
    }
}

// ---------------------------------------------------------------------------
// classification focal loss, specialized C==80 (20 float4 per anchor row).
// 64 anchors per block; codes staged into LDS via async global->LDS (CDNA5).
// ---------------------------------------------------------------------------
#define ANCH_PER_BLOCK 64

__global__ void cls_loss_80(const float* __restrict__ cls,       // [B,A,80]
                            const int* __restrict__ codes,       // [B,A]
                            const unsigned* __restrict__ nvalid, // [B]
                            float* __restrict__ accums,
                            int A) {
    constexpr int QPA = 20;    // float4s per anchor (C/4)
    __shared__ int   sCodes[ANCH_PER_BLOCK];
    __shared__ float sred[8];

    const int img  = blockIdx.y;
    const int base = blockIdx.x * ANCH_PER_BLOCK;
    const int na   = min(ANCH_PER_BLOCK, A - base);
    const int* gc  = codes + (size_t)img * A + base;

#if HAVE_ASYNC_LDS
    if (threadIdx.x < 32) {
        for (int i = (int)threadIdx.x; i < na; i += 32) {
            __builtin_amdgcn_global_load_async_to_lds_b32(
                (as1_int*)const_cast<int*>(gc + i),
                (as3_int*)(&sCodes[i]),
                0, 0);
        }
        __builtin_amdgcn_s_wait_asynccnt(0);
    }
#else
    for (int i = threadIdx.x; i < na; i += blockDim.x) sCodes[i] = gc[i];
#endif
    __syncthreads();

    const bool empty = (nvalid[img] == 0);       // all-background branch
    const int  nq    = na * QPA;
    const float4* p  = (const float4*)(cls + ((size_t)img * A + base) * 80);

    float sum = 0.0f;
    for (int q = threadIdx.x; q < nq; q += blockDim.x) {
        const float4 v = p[q];
        const int a    = q / QPA;                // compile-time divisor -> mul/shift
        const int c0   = (q - a * QPA) * 4;
        const int code = empty ? -1 : sCodes[a];
        if (code != -2) {
            sum += focal_term(v.x, c0 + 0 == code);
            sum += focal_term(v.y, c0 + 1 == code);
            sum += focal_term(v.z, c0 + 2 == code);
            sum += focal_term(v.w, c0 + 3 == code);
        }
    }
    float s = block_reduce_sum(sum, sred);
    if (threadIdx.x == 0 && s != 0.0f) atomicAdd(&accums[img], s);
}

// generic fallback for C != 80 (scalar, still coalesced)
__global__ void cls_loss_generic(const float* __restrict__ cls,
                                 const int* __restrict__ codes,
                                 const unsigned* __restrict__ nvalid,
                                 float* __restrict__ accums,
                                 int A, int C) {
    __shared__ float sred[8];
    const int img = blockIdx.y;
    const long long total = (long long)A * C;
    const bool empty = (nvalid[img] == 0);
    const float* p = cls + (long long)img * total;
    const int* cd  = codes + (long long)img * A;
    float sum = 0.0f;
    const long long start = (long long)blockIdx.x * blockDim.x * 16;
    for (int k = 0; k < 16; ++k) {
        const long long idx = start + (long long)k * blockDim.x + threadIdx.x;
        if (idx < total) {
            const int a = (int)(idx / C), c = (int)(idx - (long long)a * C);
            const int code = empty ? -1 : cd[a];
            if (code != -2) sum += focal_term(p[idx], code >= 0 && c == code);
        }
    }
    float s = block_reduce_sum(sum, sred);
    if (threadIdx.x == 0 && s != 0.0f) atomicAdd(&accums[img], s);
}

// ---------------------------------------------------------------------------
// finalize: per-image normalization + means -> out[0]=cls, out[1]=reg
// ---------------------------------------------------------------------------
__global__ void finalize_kernel(const float* __restrict__ accums,
                                const unsigned* __restrict__ nvalid,
                                float* __restrict__ out, int B) {
    if (threadIdx.x == 0 && blockIdx.x == 0) {
        float cl = 0.0f, rl = 0.0f;
        for (int i = 0; i < B; ++i) {
            const float csum = accums[i];
            const float rsum = accums[B + i];
            const float npos = accums[2 * B + i];
            const float cls  = (nvalid[i] > 0) ? csum / fmaxf(npos, 1.0f) : csum;
            const float reg  = (npos > 0.0f) ? rsum / (fmaxf(npos, 1.0f) * 4.0f) * 2.0f : 0.0f;
            cl += cls; rl += reg;
        }
        out[0] = cl / (float)B;
        out[1] = rl / (float)B;
    }
}

// ---------------------------------------------------------------------------
extern "C" void kernel_launch(void* const* d_in, const int* in_sizes, int n_in,
                              void* d_out, int out_size, void* d_ws, size_t ws_size,
                              hipStream_t stream) {
    const float* cls  = (const float*)d_in[0];   // [B,A,C]
    const float* regs = (const float*)d_in[1];   // [B,A,4]
    const float* anch = (const float*)d_in[2];   // [1,A,4]
    const float* anns = (const float*)d_in[3];   // [B,M,5]

    const int A = in_sizes[2] / 4;
    const int B = (int)((long long)in_sizes[1] / ((long long)A * 4));
    const int C = (int)((long long)in_sizes[0] / ((long long)B * A));
    const int M = (int)((long long)in_sizes[3] / ((long long)B * 5));

    int*      codes  = (int*)d_ws;                                          // B*A ints
    float*    accums = (float*)((char*)d_ws + (size_t)B * A * sizeof(int)); // 3*B floats
    unsigned* nval   = (unsigned*)(accums + 3 * B);                         // B uints

    init_kernel<<<dim3(B), dim3(64), 0, stream>>>(anns, accums, nval, B, M);

    dim3 g1((A + 255) / 256, B);
    assign_kernel<<<g1, dim3(256), 0, stream>>>(anch, regs, anns, codes, accums, A, M, B);

    if (C == 80) {
        dim3 g2((A + ANCH_PER_BLOCK - 1) / ANCH_PER_BLOCK, B);
        cls_loss_80<<<g2, dim3(256), 0, stream>>>(cls, codes, nval, accums, A);
    } else {
        const long long total = (long long)A * C;
        dim3 g2((unsigned)((total + 4095) / 4096), B);
        cls_loss_generic<<<g2, dim3(256), 0, stream>>>(cls, codes, nval, accums, A, C);
    }

    finalize_kernel<<<1, 32, 0, stream>>>(accums, nval, (float*)d_out, B);
}